// KWinners_5050881540502
// MI455X (gfx1250) — compile-verified
//
#include <hip/hip_runtime.h>
#include <stdint.h>

#define TOPK      655
#define NTHREADS  512
#define MAXN      16384
#define NWAVES    (NTHREADS / 32)

// ---------------------------------------------------------------------------
// Kernel A: boostFactors[i] = exp((K/N - dutyCycle[i]) * 1.0)
// ---------------------------------------------------------------------------
__global__ void kw_boost_kernel(const float* __restrict__ duty,
                                float* __restrict__ bf,
                                int n, float targetDensity) {
    int i = blockIdx.x * blockDim.x + threadIdx.x;
    if (i < n) bf[i] = expf(targetDensity - duty[i]);
}

__device__ __forceinline__ uint32_t kw_key(float v) {
    uint32_t b = __float_as_uint(v);
    return b ^ ((uint32_t)((int32_t)b >> 31) | 0x80000000u);
}

// ---------------------------------------------------------------------------
// Kernel B: one workgroup per row; row staged in LDS via gfx1250 async
// global->LDS copies; exact top-K threshold via 4-pass MSB radix select with
// parallel 256-bin suffix scans; barrier-free vectorized mask write.
// ---------------------------------------------------------------------------
__global__ __launch_bounds__(NTHREADS)
void kw_select_kernel(const float* __restrict__ x,
                      const float* __restrict__ bf,
                      float* __restrict__ out,
                      int N, int K) {
    __shared__ __align__(16) uint32_t s_keys[MAXN];   // 64 KiB row of keys
    __shared__ uint32_t s_hist[256];
    __shared__ uint32_t s_scan[256];
    __shared__ uint32_t s_wavecnt[NWAVES];
    __shared__ uint32_t s_bcast[3];

    const int row = blockIdx.x;
    const int tid = threadIdx.x;
    const float* xrow = x + (size_t)row * (size_t)N;

    // ---- 1) Stage row into LDS with async global->LDS (ASYNCcnt-tracked) --
    {
        uint32_t lds_base = (uint32_t)(uintptr_t)(void*)s_keys;  // LDS offset
        for (int e = tid * 4; e < N; e += NTHREADS * 4) {
            uint32_t ldsoff = lds_base + (uint32_t)e * 4u;
            uint32_t goff   = (uint32_t)e * 4u;
            asm volatile("global_load_async_to_lds_b128 %0, %1, %2"
                         :
                         : "v"(ldsoff), "v"(goff), "s"(xrow)
                         : "memory");
        }
        asm volatile("s_wait_asynccnt 0" ::: "memory");
    }
    if (tid < 256) s_hist[tid] = 0u;
    __syncthreads();

    // ---- 2) In-place transform to monotonic keys + MSB-byte histogram -----
    for (int i = tid * 4; i < N; i += NTHREADS * 4) {
        uint4  kb = *(const uint4 *)&s_keys[i];
        float4 f  = *(const float4*)&bf[i];
        uint4  k;
        k.x = kw_key(__uint_as_float(kb.x) * f.x);
        k.y = kw_key(__uint_as_float(kb.y) * f.y);
        k.z = kw_key(__uint_as_float(kb.z) * f.z);
        k.w = kw_key(__uint_as_float(kb.w) * f.w);
        *(uint4*)&s_keys[i] = k;
        atomicAdd(&s_hist[k.x >> 24], 1u);
        atomicAdd(&s_hist[k.y >> 24], 1u);
        atomicAdd(&s_hist[k.z >> 24], 1u);
        atomicAdd(&s_hist[k.w >> 24], 1u);
    }
    __syncthreads();

    // ---- 3) MSB-first radix select with parallel suffix scans -------------
    uint32_t prefix = 0u, pmask = 0u;
    uint32_t r = (uint32_t)K;          // residual rank within current prefix
    for (int s = 3; ; --s) {
        // Parallel inclusive suffix scan of s_hist into s_scan (256 bins).
        uint32_t h = 0u;
        if (tid < 256) { h = s_hist[tid]; s_scan[tid] = h; }
        __syncthreads();
        for (int off = 1; off < 256; off <<= 1) {
            uint32_t t = 0u;
            if (tid < 256) {
                t = s_scan[tid];
                if (tid + off < 256) t += s_scan[tid + off];
            }
            __syncthreads();
            if (tid < 256) s_scan[tid] = t;
            __syncthreads();
        }
        if (tid < 256) {
            uint32_t incl = s_scan[tid];   // # keys with byte >= tid (in prefix)
            uint32_t excl = incl - h;      // # keys with byte >  tid
            if (excl < r && incl >= r) {   // unique winner bin
                s_bcast[0] = (uint32_t)tid;
                s_bcast[1] = r - excl;     // residual rank inside this bin
                s_bcast[2] = h;            // population of this bin
            }
        }
        __syncthreads();
        uint32_t bin = s_bcast[0];
        r = s_bcast[1];
        prefix |= bin   << (8 * s);
        pmask  |= 0xFFu << (8 * s);
        if (s == 0) break;

        // Histogram next byte among keys matching the refined prefix.
        if (tid < 256) s_hist[tid] = 0u;
        __syncthreads();
        const int sh = 8 * (s - 1);
        for (int i = tid * 4; i < N; i += NTHREADS * 4) {
            uint4 k = *(const uint4*)&s_keys[i];
            if ((k.x & pmask) == prefix) atomicAdd(&s_hist[(k.x >> sh) & 0xFFu], 1u);
            if ((k.y & pmask) == prefix) atomicAdd(&s_hist[(k.y >> sh) & 0xFFu], 1u);
            if ((k.z & pmask) == prefix) atomicAdd(&s_hist[(k.z >> sh) & 0xFFu], 1u);
            if ((k.w & pmask) == prefix) atomicAdd(&s_hist[(k.w >> sh) & 0xFFu], 1u);
        }
        __syncthreads();
    }

    const uint32_t T        = prefix;       // K-th largest key
    const uint32_t need_eq  = r;            // equals to include (>= 1)
    const uint32_t count_eq = s_bcast[2];   // total equals (population of bin)
    float* orow = out + (size_t)row * (size_t)N;

    if (count_eq == need_eq) {
        // ---- 4a) Fast path: all threshold-equal elements are winners ------
        // mask = (key >= T); fully parallel, no barriers, b128 stores.
        for (int i = tid * 4; i < N; i += NTHREADS * 4) {
            uint4 k = *(const uint4*)&s_keys[i];
            float4 o;
            o.x = (k.x >= T) ? 1.0f : 0.0f;
            o.y = (k.y >= T) ? 1.0f : 0.0f;
            o.z = (k.z >= T) ? 1.0f : 0.0f;
            o.w = (k.w >= T) ? 1.0f : 0.0f;
            *(float4*)&orow[i] = o;
        }
    } else {
        // ---- 4b) Rare path: duplicates at threshold; stable index order ---
        const int lane = tid & 31;
        const int wid  = tid >> 5;
        uint32_t running = 0u;             // uniform across the block
        for (int c = 0; c * NTHREADS < N; ++c) {
            int i = c * NTHREADS + tid;
            uint32_t key = s_keys[i];
            bool eq = (key == T);
            uint32_t m = __builtin_amdgcn_ballot_w32(eq);   // wave32 ballot
            if (lane == 0) s_wavecnt[wid] = (uint32_t)__popc(m);
            __syncthreads();
            uint32_t base = running;
            for (int w = 0; w < NWAVES; ++w) {
                uint32_t cw = s_wavecnt[w];
                if (w < wid) base += cw;
                running += cw;
            }
            uint32_t ord = base + (uint32_t)__popc(m & ((1u << lane) - 1u));
            orow[i] = (key > T || (eq && ord < need_eq)) ? 1.0f : 0.0f;
            __syncthreads();   // protect s_wavecnt for next chunk
        }
    }
}

// ---------------------------------------------------------------------------
extern "C" void kernel_launch(void* const* d_in, const int* in_sizes, int n_in,
                              void* d_out, int out_size, void* d_ws, size_t ws_size,
                              hipStream_t stream) {
    const float* x    = (const float*)d_in[0];
    const float* duty = (const float*)d_in[1];
    float* out = (float*)d_out;
    float* bf  = (float*)d_ws;

    const int N = in_sizes[1];
    const int B = in_sizes[0] / N;
    const float targetDensity = (float)TOPK / (float)N;

    kw_boost_kernel<<<(N + 255) / 256, 256, 0, stream>>>(duty, bf, N, targetDensity);
    kw_select_kernel<<<B, NTHREADS, 0, stream>>>(x, bf, out, N, TOPK);
}